// ResNet_45183055954225
// MI455X (gfx1250) — compile-verified
//
#include <hip/hip_runtime.h>

// ---------------- problem constants ----------------
#define ANCH 69354
#define NC   80
#define TK   1000
#define KP   1024          // padded candidate count
#define SCORE_TH 0.05f
#define IOU_TH   0.5f

typedef __attribute__((ext_vector_type(2)))  float        v2f;
typedef __attribute__((ext_vector_type(8)))  float        v8f;
typedef __attribute__((ext_vector_type(4)))  unsigned int v4u;
typedef __attribute__((ext_vector_type(8)))  int          v8i;
typedef __attribute__((ext_vector_type(4)))  int          v4i;

// order-preserving float <-> uint mapping (total order, -1.0 sorts low)
__device__ __forceinline__ unsigned f2ord(float f) {
    unsigned u = __float_as_uint(f);
    return (u & 0x80000000u) ? ~u : (u | 0x80000000u);
}
__device__ __forceinline__ float ord2f(unsigned o) {
    unsigned u = (o & 0x80000000u) ? (o & 0x7FFFFFFFu) : ~o;
    return __uint_as_float(u);
}
__device__ __forceinline__ float masked_score(const float* __restrict__ cls, int a, int c) {
    float s = cls[(size_t)a * NC + c];
    return s > SCORE_TH ? s : -1.0f;
}

// ---------------- kernel 1: decode + clip ----------------
__global__ void k_decode(const float* __restrict__ reg, const float* __restrict__ anc,
                         float* __restrict__ boxes, const int* __restrict__ p_ih,
                         const int* __restrict__ p_iw) {
    int a = blockIdx.x * blockDim.x + threadIdx.x;
    if (a >= ANCH) return;
    float imh = (float)*p_ih, imw = (float)*p_iw;
    float4 an = ((const float4*)anc)[a];
    float4 dl = ((const float4*)reg)[a];
    float w = an.z - an.x, h = an.w - an.y;
    float cx = an.x + 0.5f * w, cy = an.y + 0.5f * h;
    float pcx = cx + dl.x * 0.1f * w;
    float pcy = cy + dl.y * 0.1f * h;
    float pw = expf(dl.z * 0.2f) * w;
    float ph = expf(dl.w * 0.2f) * h;
    float4 o;
    o.x = fminf(fmaxf(pcx - 0.5f * pw, 0.0f), imw);
    o.y = fminf(fmaxf(pcy - 0.5f * ph, 0.0f), imh);
    o.z = fminf(fmaxf(pcx + 0.5f * pw, 0.0f), imw);
    o.w = fminf(fmaxf(pcy + 0.5f * ph, 0.0f), imh);
    ((float4*)boxes)[a] = o;
}

// ---------------- kernel 2: per-class exact top-1000 (radix-select + bitonic) ----------------
__global__ __launch_bounds__(1024) void k_topk(const float* __restrict__ cls,
                                               const float* __restrict__ boxes,
                                               float* __restrict__ sc,
                                               float* __restrict__ bc) {
    const int c   = blockIdx.x;
    const int tid = threadIdx.x;

    __shared__ unsigned hist[256];
    __shared__ unsigned scanbuf[1024];
    __shared__ unsigned long long cand[1024];
    __shared__ unsigned s_prefix, s_need, s_gt, s_eqbase;

    if (tid == 0) { s_prefix = 0u; s_need = TK; }
    __syncthreads();

    // ---- MSB-first radix select of the TK-th largest ordered key ----
    for (int shift = 24; shift >= 0; shift -= 8) {
        if (tid < 256) hist[tid] = 0u;
        __syncthreads();
        unsigned pref   = s_prefix;
        unsigned upmask = (shift == 24) ? 0u : (0xFFFFFFFFu << (shift + 8));
        for (int a = tid; a < ANCH; a += 1024) {
            unsigned o = f2ord(masked_score(cls, a, c));
            if ((o & upmask) == (pref & upmask))
                atomicAdd(&hist[(o >> shift) & 0xFFu], 1u);
        }
        __syncthreads();
        if (tid == 0) {
            unsigned cum = 0, need = s_need, chosen = 0, above = 0;
            for (int d = 255; d >= 0; --d) {
                unsigned hh = hist[d];
                if (cum + hh >= need) { chosen = (unsigned)d; above = cum; break; }
                cum += hh;
            }
            s_prefix = pref | (chosen << shift);
            s_need   = need - above;
        }
        __syncthreads();
    }

    const unsigned T     = s_prefix;
    const unsigned quota = s_need;          // #elements == T to take (index order)
    const unsigned cntGT = TK - quota;      // #elements strictly > T

    if (tid == 0) { s_gt = 0u; s_eqbase = 0u; }
    cand[tid] = 0ull;                       // pad entries sort last
    __syncthreads();

    // ---- ordered compaction: all >T unordered, ==T in index order up to quota ----
    for (int base = 0; base < ANCH; base += 1024) {
        int a = base + tid;
        bool ok = a < ANCH;
        unsigned o = ok ? f2ord(masked_score(cls, a, c)) : 0u;
        if (ok && o > T) {
            unsigned slot = atomicAdd(&s_gt, 1u);
            cand[slot] = ((unsigned long long)o << 32) | (unsigned long long)(0xFFFFFFFFu - (unsigned)a);
        }
        unsigned p = (ok && o == T) ? 1u : 0u;
        scanbuf[tid] = p;
        __syncthreads();
        for (int off = 1; off < 1024; off <<= 1) {          // inclusive Hillis-Steele scan
            unsigned v = (tid >= off) ? scanbuf[tid - off] : 0u;
            __syncthreads();
            scanbuf[tid] += v;
            __syncthreads();
        }
        unsigned incl  = scanbuf[tid];
        unsigned total = scanbuf[1023];
        unsigned eb    = s_eqbase;
        if (p && (eb + (incl - p)) < quota) {
            unsigned slot = cntGT + eb + (incl - p);
            cand[slot] = ((unsigned long long)o << 32) | (unsigned long long)(0xFFFFFFFFu - (unsigned)a);
        }
        __syncthreads();
        if (tid == 0) s_eqbase = eb + total;
        __syncthreads();
    }

    // ---- bitonic sort, descending on (ordered score, ~index) ----
    for (unsigned kk = 2; kk <= 1024; kk <<= 1) {
        for (unsigned j = kk >> 1; j > 0; j >>= 1) {
            __syncthreads();
            unsigned ixj = (unsigned)tid ^ j;
            if (ixj > (unsigned)tid) {
                unsigned long long x = cand[tid], y = cand[ixj];
                bool desc = ((tid & kk) == 0);
                if (desc ? (x < y) : (x > y)) { cand[tid] = y; cand[ixj] = x; }
            }
        }
    }
    __syncthreads();

    // ---- emit sorted scores + gathered boxes (padded to KP) ----
    unsigned long long key = cand[tid];
    unsigned o   = (unsigned)(key >> 32);
    unsigned idx = 0xFFFFFFFFu - (unsigned)(key & 0xFFFFFFFFull);
    sc[(size_t)c * KP + tid] = (tid < TK) ? ord2f(o) : -1.0f;
    float4 b = make_float4(0.f, 0.f, 0.f, 0.f);
    if (tid < TK && idx < (unsigned)ANCH) b = ((const float4*)boxes)[idx];
    ((float4*)bc)[(size_t)c * KP + tid] = b;
}

// ---------------- kernel 3: IoU suppression bit-matrix (WMMA + TDM) ----------------
__global__ __launch_bounds__(256) void k_iou_bits(const float* __restrict__ bc,
                                                  unsigned* __restrict__ bitmat) {
    const int c    = blockIdx.y;
    const int tr   = blockIdx.x;           // row tile: rows tr*16 .. tr*16+15
    const int tid  = threadIdx.x;
    const int wave = tid >> 5;
    const int lane = tid & 31;

    __shared__ __align__(16) float smem[KP * 4];   // 16 KB of class boxes
    __shared__ float sarea[KP];

    // stage this class's 1024 boxes (16 KB) into LDS via the Tensor Data Mover
    const float* src = bc + (size_t)c * KP * 4;
    if (wave == 0) {
        unsigned lds_off = (unsigned)(unsigned long long)(&smem[0]);
        unsigned long long ga = (unsigned long long)(const void*)src;
        v4u g0;
        g0.x = 1u;                                              // count = 1 valid descriptor
        g0.y = lds_off;                                         // lds_addr
        g0.z = (unsigned)(ga & 0xFFFFFFFFull);                  // global_addr[31:0]
        g0.w = (unsigned)((ga >> 32) & 0x1FFFFFFull) | (2u << 30); // global_addr[56:32], type=2
        v8i g1;
        g1[0] = (int)(2u << 16);        // data_size = 4 bytes
        g1[1] = (int)(4096u << 16);     // tensor_dim0[15:0]=4096 -> bits[63:48]
        g1[2] = (int)(1u << 16);        // tensor_dim0 hi=0 | tensor_dim1[15:0]=1
        g1[3] = (int)(4096u << 16);     // tensor_dim1 hi=0 | tile_dim0=4096
        g1[4] = 1;                      // tile_dim1=1, tile_dim2=0
        g1[5] = 4096;                   // tensor_dim0_stride lo
        g1[6] = 0;
        g1[7] = 0;
        v4i g2 = {0, 0, 0, 0};          // group 2 unused (2D tensor)
        v4i g3 = {0, 0, 0, 0};          // group 3 unused
        v8i g4 = {0, 0, 0, 0, 0, 0, 0, 0};
        __builtin_amdgcn_tensor_load_to_lds(g0, g1, g2, g3, g4, 0);
        __builtin_amdgcn_s_wait_tensorcnt(0);
    }
    __syncthreads();

    for (int i = tid; i < KP; i += 256)
        sarea[i] = (smem[i * 4 + 2] - smem[i * 4]) * (smem[i * 4 + 3] - smem[i * 4 + 1]);
    __syncthreads();

    const int r0 = tr * 16;
    const int hi = lane >> 4;
    const int nl = lane & 15;
    const float lo16 = (lane < 16) ? 1.0f : 0.0f;   // select mask, no exec divergence

    // A fragment (16x4 f32): row M = lane (lanes 0-15), K=0 -> area, K=1 -> 1, K=2,3 -> 0
    float arow = sarea[r0 + nl];                     // unconditional LDS load
    v2f afrag;
    afrag.x = arow * lo16;      // lanes 16-31 carry K=2 -> 0
    afrag.y = lo16;             // K=1 -> 1 (lanes 0-15), K=3 -> 0

    for (int pairIdx = wave; pairIdx < 32; pairIdx += 8) {
        const int c0 = pairIdx * 32;
        // B fragments (4x16 f32): col N = lane (lanes 0-15), K=0 -> 1, K=1 -> area_col
        float ac0 = sarea[c0 + nl];
        float ac1 = sarea[c0 + 16 + nl];
        v2f b0, b1;
        b0.x = lo16;  b0.y = ac0 * lo16;
        b1.x = lo16;  b1.y = ac1 * lo16;
        v8f z = {0.f, 0.f, 0.f, 0.f, 0.f, 0.f, 0.f, 0.f};
        // D[M][N] = area_row[M] + area_col[N]  (pairwise area-sum via matrix engine)
        v8f d0 = __builtin_amdgcn_wmma_f32_16x16x4_f32(false, afrag, false, b0, (short)0, z, false, false);
        v8f d1 = __builtin_amdgcn_wmma_f32_16x16x4_f32(false, afrag, false, b1, (short)0, z, false, false);

        float4 cb0 = ((const float4*)smem)[c0 + nl];
        float4 cb1 = ((const float4*)smem)[c0 + 16 + nl];

#pragma unroll
        for (int v = 0; v < 8; ++v) {
            int m = r0 + v + 8 * hi;               // CDNA5 C/D layout: VGPR v, lane half -> row
            float4 rb = ((const float4*)smem)[m];
            float iw0 = fmaxf(fminf(rb.z, cb0.z) - fmaxf(rb.x, cb0.x), 0.f);
            float ih0 = fmaxf(fminf(rb.w, cb0.w) - fmaxf(rb.y, cb0.y), 0.f);
            float in0 = iw0 * ih0;
            float iw1 = fmaxf(fminf(rb.z, cb1.z) - fmaxf(rb.x, cb1.x), 0.f);
            float ih1 = fmaxf(fminf(rb.w, cb1.w) - fmaxf(rb.y, cb1.y), 0.f);
            float in1 = iw1 * ih1;
            // iou > th  <=>  inter > th*(areaSum - inter + eps)   (denominator > 0)
            bool p0 = in0 > IOU_TH * (d0[v] - in0 + 1e-8f);
            bool p1 = in1 > IOU_TH * (d1[v] - in1 + 1e-8f);
            unsigned m0 = (unsigned)__ballot(p0);   // [15:0]=row v, [31:16]=row v+8
            unsigned m1 = (unsigned)__ballot(p1);
            unsigned wlow  = (m0 & 0xFFFFu) | ((m1 & 0xFFFFu) << 16);
            unsigned whigh = (m0 >> 16) | (m1 & 0xFFFF0000u);
            if (lane == 0) bitmat[((size_t)c * KP + (r0 + v)) * 32 + pairIdx] = wlow;
            if (lane == 1) bitmat[((size_t)c * KP + (r0 + 8 + v)) * 32 + pairIdx] = whigh;
        }
    }
}

// ---------------- kernel 4: single-wave greedy scan + output format ----------------
__global__ __launch_bounds__(1024) void k_nms_out(const float* __restrict__ sc,
                                                  const float* __restrict__ bc,
                                                  const unsigned* __restrict__ bitmat,
                                                  float* __restrict__ out,
                                                  int* __restrict__ out_cls) {
    const int c   = blockIdx.x;
    const int tid = threadIdx.x;
    __shared__ unsigned keepbuf[32];

    if (tid < 32) {
        const int lane = tid;
        unsigned keep = 0u;
        for (int b = 0; b < 32; ++b) {
            int k = lane * 32 + b;
            if (k < TK && sc[(size_t)c * KP + k] > SCORE_TH) keep |= (1u << b);
        }
        const unsigned* bm = bitmat + (size_t)c * KP * 32;
        for (int i = 0; i < TK; ++i) {
            int owner = i >> 5, bit = i & 31;
            if (i + 1 < TK) __builtin_prefetch(&bm[(size_t)(i + 1) * 32 + lane], 0, 1);
            unsigned kw = __shfl(keep, owner, 32);
            if ((kw >> bit) & 1u) {
                unsigned sup = bm[(size_t)i * 32 + lane];
                if (lane < owner)       sup = 0u;
                else if (lane == owner) sup &= (bit == 31) ? 0u : (0xFFFFFFFFu << (bit + 1));
                keep &= ~sup;
            }
        }
        keepbuf[lane] = keep;
    }
    __syncthreads();

    if (tid < TK) {
        bool kept = (keepbuf[tid >> 5] >> (tid & 31)) & 1u;
        float s = sc[(size_t)c * KP + tid];
        float4 b4 = ((const float4*)bc)[(size_t)c * KP + tid];
        size_t row = (size_t)c * TK + tid;
        float* o = out + row * 5;
        o[0] = kept ? s    : -1.0f;
        o[1] = kept ? b4.x : 0.0f;
        o[2] = kept ? b4.y : 0.0f;
        o[3] = kept ? b4.z : 0.0f;
        o[4] = kept ? b4.w : 0.0f;
        out_cls[row] = c;
    }
}

// ---------------- launch ----------------
extern "C" void kernel_launch(void* const* d_in, const int* in_sizes, int n_in,
                              void* d_out, int out_size, void* d_ws, size_t ws_size,
                              hipStream_t stream) {
    (void)in_sizes; (void)n_in; (void)out_size; (void)ws_size;
    const float* cls = (const float*)d_in[0];
    const float* reg = (const float*)d_in[1];
    const float* anc = (const float*)d_in[2];
    const int*   ih  = (const int*)d_in[3];
    const int*   iw  = (const int*)d_in[4];

    char* ws = (char*)d_ws;
    size_t o_boxes = 0;
    size_t o_bc = (o_boxes + (size_t)ANCH * 16 + 255) & ~(size_t)255;
    size_t o_sc = (o_bc + (size_t)NC * KP * 16 + 255) & ~(size_t)255;
    size_t o_bm = (o_sc + (size_t)NC * KP * 4 + 255) & ~(size_t)255;
    float*    boxes  = (float*)(ws + o_boxes);
    float*    bc     = (float*)(ws + o_bc);
    float*    sc     = (float*)(ws + o_sc);
    unsigned* bitmat = (unsigned*)(ws + o_bm);

    k_decode<<<(ANCH + 255) / 256, 256, 0, stream>>>(reg, anc, boxes, ih, iw);
    k_topk<<<NC, 1024, 0, stream>>>(cls, boxes, sc, bc);
    dim3 g3(KP / 16, NC);
    k_iou_bits<<<g3, 256, 0, stream>>>(bc, bitmat);
    float* det = (float*)d_out;
    int*   cid = (int*)(det + (size_t)NC * TK * 5);
    k_nms_out<<<NC, 1024, 0, stream>>>(sc, bc, bitmat, det, cid);
}